// SVX_90263032693227
// MI455X (gfx1250) — compile-verified
//
#include <hip/hip_runtime.h>
#include <hip/hip_bf16.h>

typedef __attribute__((ext_vector_type(2))) float v2f;
typedef __attribute__((ext_vector_type(8))) float v8f;

#define L_DIM 8
#define H_DIM 256
#define W_DIM 448
#define KL 2
#define KH 18
#define KW 32
#define KHW (KH*KW)
#define KSP (KL*KHW)            /* 1152 superpixels */
#define NPIX (L_DIM*H_DIM*W_DIM)/* 917504 pixels    */
#define NW 4                    /* waves per cell block */
#define CELL_THREADS (NW*32)

// scales (match reference python-double -> f32 cast)
static constexpr float TSC  = (float)(2.0 / (0.4 * 8.0));      // 0.625
static constexpr float YXA  = (float)(18.0 / (0.4 * 256.0));
static constexpr float YXB  = (float)(32.0 / (0.4 * 448.0));
static constexpr float YXSC = (YXA > YXB) ? YXA : YXB;         // 0.1785714...
static constexpr float LABSC = 0.26f;

static __device__ __forceinline__ v8f wmma4(v2f a, v2f b, v8f c) {
  // D(16x16,f32) = A(16x4,f32) * B(4x16,f32) + C  -> v_wmma_f32_16x16x4_f32
  return __builtin_amdgcn_wmma_f32_16x16x4_f32(false, a, false, b, (short)0, c, false, false);
}

// -------- pFeat: write 6 channels (also serves as the feature buffer) -------
__global__ void prep_kernel(const float* __restrict__ lab, float* __restrict__ outPF) {
  int p = blockIdx.x * blockDim.x + threadIdx.x;
  if (p >= NPIX) return;
  int t = p / (H_DIM * W_DIM);
  int r = p - t * (H_DIM * W_DIM);
  int y = r / W_DIM;
  int x = r - y * W_DIM;
  outPF[0 * NPIX + p] = TSC  * (float)t;
  outPF[1 * NPIX + p] = YXSC * (float)y;
  outPF[2 * NPIX + p] = YXSC * (float)x;
  outPF[3 * NPIX + p] = LABSC * lab[0 * NPIX + p];
  outPF[4 * NPIX + p] = LABSC * lab[1 * NPIX + p];
  outPF[5 * NPIX + p] = LABSC * lab[2 * NPIX + p];
}

__global__ void zero_kernel(float* __restrict__ a, int n) {
  int i = blockIdx.x * blockDim.x + threadIdx.x;
  if (i < n) a[i] = 0.f;
}

// -------- scatter-mean init: atomic accumulate (f,1) by input idx ----------
__global__ void init_accum_kernel(const float* __restrict__ pF,
                                  const int* __restrict__ idx,
                                  float* __restrict__ spAcc) {
  int p = blockIdx.x * blockDim.x + threadIdx.x;
  if (p >= NPIX) return;
  int k = idx[p];
  if ((unsigned)k >= (unsigned)KSP) return;
  #pragma unroll
  for (int c = 0; c < 6; ++c) atomicAdd(spAcc + k * 8 + c, pF[c * NPIX + p]);
  atomicAdd(spAcc + k * 8 + 6, 1.0f);
}

// -------- normalize accumulators -> raw sp features + |g|^2 ----------------
__global__ void normalize_kernel(const float* __restrict__ spAcc,
                                 float* __restrict__ spRaw, float eps) {
  int k = blockIdx.x * blockDim.x + threadIdx.x;
  if (k >= KSP) return;
  float w  = spAcc[k * 8 + 6];
  float wm = fmaxf(w, eps);
  float n2 = 0.f;
  #pragma unroll
  for (int c = 0; c < 6; ++c) {
    float g = spAcc[k * 8 + c] / wm;
    spRaw[k * 8 + c] = g;
    n2 += g * g;
  }
  spRaw[k * 8 + 6] = n2;
  spRaw[k * 8 + 7] = 0.f;
}

// -------- fused assoc (+update accumulate, or +final outputs) per cell -----
template <bool LAST>
__global__ __launch_bounds__(CELL_THREADS, 1)
void cell_kernel(const float* __restrict__ pF,
                 const float* __restrict__ spRaw,
                 float* __restrict__ spAcc,
                 float* __restrict__ assocOut,
                 float* __restrict__ finalOut) {
  const int cell = blockIdx.x;
  const int kl = cell / KHW;
  const int kh = (cell - kl * KHW) / KW;
  const int kw = cell % KW;

  // pixel extents of this grid cell ( (v*Kd)//D == k  <=>  v in [ceil, ceil) )
  const int t0 = (kl * L_DIM + KL - 1) / KL, t1 = ((kl + 1) * L_DIM + KL - 1) / KL;
  const int y0 = (kh * H_DIM + KH - 1) / KH, y1 = ((kh + 1) * H_DIM + KH - 1) / KH;
  const int x0 = (kw * W_DIM + KW - 1) / KW, x1 = ((kw + 1) * W_DIM + KW - 1) / KW;
  const int ct = t1 - t0, cy = y1 - y0, cx = x1 - x0;
  const int npix = ct * cy * cx;

  __shared__ float gS[32 * 8];   // rows: Ghat_j = (-2g0..-2g5, 1, |g|^2), rows 27..31 zero
  __shared__ int   nidxS[27];

  const int tid = threadIdx.x;
  if (tid < 32) {
    float gv[8] = {0.f, 0.f, 0.f, 0.f, 0.f, 0.f, 0.f, 0.f};
    if (tid < 27) {
      int dl = tid / 9 - 1, dh = (tid % 9) / 3 - 1, dw = tid % 3 - 1;
      int nl = kl + dl, nh = kh + dh, nw = kw + dw;
      bool v = (nl >= 0) && (nl < KL) && (nh >= 0) && (nh < KH) && (nw >= 0) && (nw < KW);
      int cl = min(max(nl, 0), KL - 1);
      int ch = min(max(nh, 0), KH - 1);
      int cw = min(max(nw, 0), KW - 1);
      int nj = cl * KHW + ch * KW + cw;
      nidxS[tid] = nj;
      if (v) {
        const float* g = spRaw + nj * 8;
        #pragma unroll
        for (int c = 0; c < 6; ++c) gv[c] = -2.f * g[c];
        gv[6] = 1.f;
        gv[7] = g[6];  // |g|^2
      }
    }
    #pragma unroll
    for (int c = 0; c < 8; ++c) gS[tid * 8 + c] = gv[c];
  }
  __syncthreads();

  // 27-bit validity mask (uniform scalar math)
  unsigned vm = 0;
  #pragma unroll
  for (int s = 0; s < 27; ++s) {
    int dl = s / 9 - 1, dh = (s % 9) / 3 - 1, dw = s % 3 - 1;
    int nl = kl + dl, nh = kh + dh, nw = kw + dw;
    if ((nl >= 0) && (nl < KL) && (nh >= 0) && (nh < KH) && (nw >= 0) && (nw < KW))
      vm |= (1u << s);
  }

  const int  lane = tid & 31;
  const bool hih  = lane >= 16;
  const int  m    = lane & 15;
  const int  base = hih ? 8 : 0;

  // A fragments (f32 16x4): lanes 0-15 hold K={0,1}, lanes 16-31 K={2,3}
  const float* g0 = gS + m * 8;          // neighbor rows 0..15
  const float* g1 = gS + (16 + m) * 8;   // neighbor rows 16..31
  const int o0 = hih ? 2 : 0;
  const int o1 = hih ? 6 : 4;
  v2f a0lo; a0lo[0] = g0[o0]; a0lo[1] = g0[o0 + 1];
  v2f a0hi; a0hi[0] = g0[o1]; a0hi[1] = g0[o1 + 1];
  v2f a1lo; a1lo[0] = g1[o0]; a1lo[1] = g1[o0 + 1];
  v2f a1hi; a1hi[0] = g1[o1]; a1hi[1] = g1[o1 + 1];

  bool vS[16];
  int  jS[16];
  #pragma unroll
  for (int s = 0; s < 16; ++s) {
    int j = ((s >> 3) << 4) + (s & 7) + base;   // slot -> global neighbor row
    jS[s] = j;
    vS[s] = (j < 27) && ((vm >> j) & 1u);
  }

  float acc[16][7];
  if (!LAST) {
    #pragma unroll
    for (int s = 0; s < 16; ++s)
      #pragma unroll
      for (int c = 0; c < 7; ++c) acc[s][c] = 0.f;
  }

  const int wave = tid >> 5;
  const int nb   = (npix + 15) >> 4;
  const int cyx  = cy * cx;

  for (int b = wave; b < nb; b += NW) {
    int  q   = b * 16 + m;
    bool pok = q < npix;
    int  qq  = pok ? q : 0;
    int  tt  = qq / cyx;  int rem = qq - tt * cyx;
    int  yy  = rem / cx;  int xx  = rem - yy * cx;
    int  p   = ((t0 + tt) * H_DIM + (y0 + yy)) * W_DIM + (x0 + xx);

    float f0 = pF[0 * NPIX + p], f1 = pF[1 * NPIX + p], f2 = pF[2 * NPIX + p];
    float f3 = pF[3 * NPIX + p], f4 = pF[4 * NPIX + p], f5 = pF[5 * NPIX + p];
    float fn2 = f0 * f0 + f1 * f1 + f2 * f2 + f3 * f3 + f4 * f4 + f5 * f5;

    // B fragment (Fhat = f0..f5, |f|^2, 1): lanes0-15 K={0,1},{4,5}; lanes16-31 K={2,3},{6,7}
    v2f blo, bhi;
    blo[0] = hih ? f2  : f0;   blo[1] = hih ? f3   : f1;
    bhi[0] = hih ? fn2 : f4;   bhi[1] = hih ? 1.0f : f5;

    v8f z = {0.f, 0.f, 0.f, 0.f, 0.f, 0.f, 0.f, 0.f};
    v8f dlo = wmma4(a0lo, blo, z);   dlo = wmma4(a0hi, bhi, dlo);   // rows 0..15
    v8f dhi = wmma4(a1lo, blo, z);   dhi = wmma4(a1hi, bhi, dhi);   // rows 16..31

    // masked softmax(-d) over 27 neighbors: in-lane 16 + one xor-16 combine
    float xv[16];
    #pragma unroll
    for (int s = 0; s < 16; ++s) {
      float dv = (s < 8) ? dlo[s & 7] : dhi[s & 7];
      xv[s] = vS[s] ? -dv : -1.0e10f;
    }
    float mloc = xv[0];
    #pragma unroll
    for (int s = 1; s < 16; ++s) mloc = fmaxf(mloc, xv[s]);
    float mo = fmaxf(mloc, __shfl_xor(mloc, 16, 32));
    float es[16]; float ssum = 0.f;
    #pragma unroll
    for (int s = 0; s < 16; ++s) { es[s] = __expf(xv[s] - mo); ssum += es[s]; }
    ssum += __shfl_xor(ssum, 16, 32);
    float inv = 1.0f / ssum;
    float av[16];
    #pragma unroll
    for (int s = 0; s < 16; ++s) av[s] = (vS[s] && pok) ? es[s] * inv : 0.f;

    if (!LAST) {
      #pragma unroll
      for (int s = 0; s < 16; ++s) {
        float a = av[s];
        acc[s][0] += a * f0; acc[s][1] += a * f1; acc[s][2] += a * f2;
        acc[s][3] += a * f3; acc[s][4] += a * f4; acc[s][5] += a * f5;
        acc[s][6] += a;
      }
    } else {
      #pragma unroll
      for (int s = 0; s < 16; ++s)
        if (jS[s] < 27 && pok) assocOut[jS[s] * NPIX + p] = av[s];
      // argmax over 27 (first-max tie-break by smallest j)
      float ba = -1.f; int bj = 0;
      #pragma unroll
      for (int s = 0; s < 16; ++s) { if (av[s] > ba) { ba = av[s]; bj = jS[s]; } }
      float ba2 = __shfl_xor(ba, 16, 32);
      int   bj2 = __shfl_xor(bj, 16, 32);
      if ((ba2 > ba) || ((ba2 == ba) && (bj2 < bj))) { ba = ba2; bj = bj2; }
      if (!hih && pok) finalOut[p] = (float)nidxS[bj];
    }
  }

  if (!LAST) {
    // reduce per-lane accumulators over the 16 pixel-lanes of each half-wave
    #pragma unroll
    for (int s = 0; s < 16; ++s)
      #pragma unroll
      for (int c = 0; c < 7; ++c) {
        float v = acc[s][c];
        v += __shfl_xor(v, 1, 32);
        v += __shfl_xor(v, 2, 32);
        v += __shfl_xor(v, 4, 32);
        v += __shfl_xor(v, 8, 32);
        acc[s][c] = v;
      }
    // lanes s and s+16 flush slot s (different j rows) with global atomics
    #pragma unroll
    for (int s = 0; s < 16; ++s) {
      if (m == s && vS[s]) {
        int kk = nidxS[jS[s]];
        #pragma unroll
        for (int c = 0; c < 7; ++c) atomicAdd(spAcc + kk * 8 + c, acc[s][c]);
      }
    }
  }
}

// -------- spFeat output: (K,6) -> (6,K) ------------------------------------
__global__ void spfeat_out_kernel(const float* __restrict__ spRaw, float* __restrict__ out) {
  int i = blockIdx.x * blockDim.x + threadIdx.x;
  if (i >= 6 * KSP) return;
  int c = i / KSP, k = i - c * KSP;
  out[i] = spRaw[k * 8 + c];
}

extern "C" void kernel_launch(void* const* d_in, const int* in_sizes, int n_in,
                              void* d_out, int out_size, void* d_ws, size_t ws_size,
                              hipStream_t stream) {
  (void)in_sizes; (void)n_in; (void)out_size; (void)ws_size;
  const float* lab = (const float*)d_in[0];
  const int*   idx = (const int*)d_in[1];

  float* outF     = (float*)d_out;
  float* pF       = outF;                  // 6*NPIX   (pFeat5 output, reused as feature buf)
  float* spOut    = outF + 6 * NPIX;       // 6*KSP
  float* assocOut = spOut + 6 * KSP;       // 27*NPIX
  float* finalOut = assocOut + 27 * NPIX;  // NPIX

  float* spRaw = (float*)d_ws;             // KSP*8 floats
  float* spAcc = spRaw + KSP * 8;          // KSP*8 floats

  dim3 b256(256);
  prep_kernel<<<(NPIX + 255) / 256, b256, 0, stream>>>(lab, pF);
  zero_kernel<<<(KSP * 8 + 255) / 256, b256, 0, stream>>>(spAcc, KSP * 8);
  init_accum_kernel<<<(NPIX + 255) / 256, b256, 0, stream>>>(pF, idx, spAcc);
  normalize_kernel<<<(KSP + 255) / 256, b256, 0, stream>>>(spAcc, spRaw, 1.0f);

  for (int it = 0; it < 4; ++it) {
    zero_kernel<<<(KSP * 8 + 255) / 256, b256, 0, stream>>>(spAcc, KSP * 8);
    cell_kernel<false><<<KSP, CELL_THREADS, 0, stream>>>(pF, spRaw, spAcc, assocOut, finalOut);
    normalize_kernel<<<(KSP + 255) / 256, b256, 0, stream>>>(spAcc, spRaw, 1e-12f);
  }
  cell_kernel<true><<<KSP, CELL_THREADS, 0, stream>>>(pF, spRaw, spAcc, assocOut, finalOut);
  spfeat_out_kernel<<<(6 * KSP + 255) / 256, b256, 0, stream>>>(spRaw, spOut);
}